// MGL_36129264894273
// MI455X (gfx1250) — compile-verified
//
#include <hip/hip_runtime.h>

// ---------------------------------------------------------------------------
// MI455X (gfx1250) fused 3x3 local-attention block.
//
// Stage 1: three 1x1 convs (64x64 GEMM) + ReLU  -> V_WMMA_F32_16X16X4_F32
//          (fp32 WMMA keeps bit-faithful precision vs fp32 reference).
// Stage 2: per-pixel 3x3 attention (9x9 inner dims) -> VALU fp32.
//
// Roofline: ~3.5 GFLOP total vs ~47 MB of unique traffic -> compute bound at
// 23.3 TB/s HBM; intermediates (57 MB) live entirely in the 192 MB L2.
// ---------------------------------------------------------------------------

typedef __attribute__((ext_vector_type(2))) float v2f;
typedef __attribute__((ext_vector_type(8))) float v8f;

#define CN        64
#define HH        96
#define WW        96
#define HWP       (HH * WW)        // 9216 pixels per image
#define IMG       (CN * HWP)       // 589824 floats per image (NCHW)
#define NPIX      (8 * HWP)        // 73728 total pixels

// ===========================================================================
// Kernel 1: D(16 outch x 16 pix) = W(16x4) x X(4x16), K=64 in chunks of 4,
// for 3 weight matrices simultaneously (B operand reused 12x per K-chunk).
//
// WMMA fp32 operand layout (ISA 7.12.2):
//   A 16x4:  vgpr0: lanes0-15 K=0 / lanes16-31 K=2 ; vgpr1: K=1 / K=3 ; M=lane&15
//   B 4x16:  mirrored (lanes hold N, K split across {vgpr, lane-half})
//   D 16x16: vgpr r: lanes0-15 -> (M=r, N=lane) ; lanes16-31 -> (M=r+8, N=lane-16)
// With M = out-channel and N = pixel, each D vgpr stores two 64B-contiguous
// pixel runs -> coalesced NCHW stores.
// ===========================================================================
__global__ __launch_bounds__(256) void conv1x1x3_wmma_relu(
    const float* __restrict__ x,
    const float* __restrict__ w1,
    const float* __restrict__ w2,
    const float* __restrict__ w3,
    float* __restrict__ y1,
    float* __restrict__ y2,
    float* __restrict__ y3)
{
    // Weights pre-swizzled into WMMA A-fragment order:
    // [wi][kc][mt][lane*2 + j]  -> per-lane ds_load_b64, conflict free.
    __shared__ float wlds[3 * 16 * 4 * 64];   // 12288 floats = 48 KB

    const int tid = threadIdx.x;
    for (int idx = tid; idx < 3 * 16 * 4 * 64; idx += 256) {
        const int j  = idx & 1;
        const int L  = (idx >> 1) & 31;
        const int mt = (idx >> 6) & 3;
        const int kc = (idx >> 8) & 15;
        const int wi = idx >> 12;
        const int o  = mt * 16 + (L & 15);           // out-channel (M)
        const int c  = 4 * kc + 2 * (L >> 4) + j;    // in-channel  (K)
        const float* wsrc = (wi == 0) ? w1 : (wi == 1) ? w2 : w3;
        wlds[idx] = wsrc[o * CN + c];
    }
    __syncthreads();

    const int lane = tid & 31;
    const int wave = tid >> 5;
    const int pix0 = blockIdx.x * 128 + wave * 16;   // 16 consecutive pixels
    const int n    = pix0 / HWP;                     // 9216 % 128 == 0: no straddle
    const int p0   = pix0 - n * HWP;

    // B operand: lane -> pixel (lane&15), K offset 2*(lane>>4) within chunk.
    const float* xb  = x + (size_t)n * IMG + p0 + (lane & 15);
    const int   koff = 2 * (lane >> 4);

    v8f acc[3][4];
#pragma unroll
    for (int wi = 0; wi < 3; ++wi)
#pragma unroll
        for (int mt = 0; mt < 4; ++mt)
            acc[wi][mt] = {};

#pragma unroll 4
    for (int kc = 0; kc < 16; ++kc) {
        const int cb = 4 * kc + koff;
        v2f b;
        b.x = xb[(size_t)cb * HWP];          // channel plane cb   (coalesced)
        b.y = xb[(size_t)(cb + 1) * HWP];    // channel plane cb+1 (coalesced)
#pragma unroll
        for (int wi = 0; wi < 3; ++wi) {
#pragma unroll
            for (int mt = 0; mt < 4; ++mt) {
                const v2f a = *(const v2f*)&wlds[(((wi * 16 + kc) * 4 + mt) << 6) + lane * 2];
                acc[wi][mt] = __builtin_amdgcn_wmma_f32_16x16x4_f32(
                    /*neg_a=*/false, a, /*neg_b=*/false, b,
                    /*c_mod=*/(short)0, acc[wi][mt],
                    /*reuse_a=*/false, /*reuse_b=*/false);
            }
        }
    }

    // Epilogue: ReLU + coalesced NCHW store.
#pragma unroll
    for (int wi = 0; wi < 3; ++wi) {
        float* yo = (wi == 0) ? y1 : (wi == 1) ? y2 : y3;
        float* yb = yo + (size_t)n * IMG + p0 + (lane & 15);
#pragma unroll
        for (int mt = 0; mt < 4; ++mt) {
#pragma unroll
            for (int r = 0; r < 8; ++r) {
                const int o = mt * 16 + r + 8 * (lane >> 4);
                const float v = acc[wi][mt][r];
                yb[(size_t)o * HWP] = v > 0.0f ? v : 0.0f;
            }
        }
    }
}

// ===========================================================================
// Kernel 2: per-pixel 3x3 attention. One thread per pixel; lanes map to
// consecutive w -> all neighbor loads coalesce. Borders are zero-padded via
// EXEC-masked loads. x5 (9x64) is recomputed per streaming pass instead of
// spilling 576 floats.
// ===========================================================================
__global__ __launch_bounds__(256) void attention3x3(
    const float* __restrict__ x1,
    const float* __restrict__ x2,
    const float* __restrict__ x3,
    float* __restrict__ out)
{
    const int p   = blockIdx.x * 256 + threadIdx.x;   // 0..73727
    const int n   = p / HWP;
    const int rem = p - n * HWP;
    const int h   = rem / WW;
    const int w   = rem - h * WW;
    const int base = n * IMG + rem;                   // < 2^23, fits int

    int  off[9];
    bool ok[9];
#pragma unroll
    for (int i = 0; i < 3; ++i)
#pragma unroll
        for (int j = 0; j < 3; ++j) {
            const int k  = i * 3 + j;
            const int dy = i - 1, dx = j - 1;
            off[k] = dy * WW + dx;
            ok[k]  = ((unsigned)(h + dy) < (unsigned)HH) &&
                     ((unsigned)(w + dx) < (unsigned)WW);
        }

    // ---- score1[k][l] = sum_c x1nb[k][c] * x2nb[l][c] -------------------
    float s[9][9];
#pragma unroll
    for (int k = 0; k < 9; ++k)
#pragma unroll
        for (int l = 0; l < 9; ++l) s[k][l] = 0.0f;

    for (int c = 0; c < CN; ++c) {
        const int cb = base + c * HWP;
        float a[9], b[9];
#pragma unroll
        for (int k = 0; k < 9; ++k) a[k] = ok[k] ? x1[cb + off[k]] : 0.0f;
#pragma unroll
        for (int l = 0; l < 9; ++l) b[l] = ok[l] ? x2[cb + off[l]] : 0.0f;
#pragma unroll
        for (int k = 0; k < 9; ++k)
#pragma unroll
            for (int l = 0; l < 9; ++l) s[k][l] = fmaf(a[k], b[l], s[k][l]);
    }

    // row softmax over l
#pragma unroll
    for (int k = 0; k < 9; ++k) {
        float m = s[k][0];
#pragma unroll
        for (int l = 1; l < 9; ++l) m = fmaxf(m, s[k][l]);
        float sum = 0.0f;
#pragma unroll
        for (int l = 0; l < 9; ++l) { s[k][l] = expf(s[k][l] - m); sum += s[k][l]; }
        const float inv = 1.0f / sum;
#pragma unroll
        for (int l = 0; l < 9; ++l) s[k][l] *= inv;
    }

    // ---- pass B: dot[k] = sum_c x5[k][c] * x5[4][c] ---------------------
    float dot[9];
#pragma unroll
    for (int k = 0; k < 9; ++k) dot[k] = 0.0f;

    for (int c = 0; c < CN; ++c) {
        const int cb = base + c * HWP;
        float t[9];
#pragma unroll
        for (int l = 0; l < 9; ++l) t[l] = ok[l] ? x3[cb + off[l]] : 0.0f;
        float y[9];
#pragma unroll
        for (int k = 0; k < 9; ++k) {
            float acc = 0.0f;
#pragma unroll
            for (int l = 0; l < 9; ++l) acc = fmaf(s[k][l], t[l], acc);
            y[k] = acc;
        }
#pragma unroll
        for (int k = 0; k < 9; ++k) dot[k] = fmaf(y[k], y[4], dot[k]);
    }

    // softmax over k -> score4
    {
        float m = dot[0];
#pragma unroll
        for (int k = 1; k < 9; ++k) m = fmaxf(m, dot[k]);
        float sum = 0.0f;
#pragma unroll
        for (int k = 0; k < 9; ++k) { dot[k] = expf(dot[k] - m); sum += dot[k]; }
        const float inv = 1.0f / sum;
#pragma unroll
        for (int k = 0; k < 9; ++k) dot[k] *= inv;
    }

    // ---- pass C: out[c] = sum_k x5[k][c] * score4[k] --------------------
    for (int c = 0; c < CN; ++c) {
        const int cb = base + c * HWP;
        float t[9];
#pragma unroll
        for (int l = 0; l < 9; ++l) t[l] = ok[l] ? x3[cb + off[l]] : 0.0f;
        float o = 0.0f;
#pragma unroll
        for (int k = 0; k < 9; ++k) {
            float acc = 0.0f;
#pragma unroll
            for (int l = 0; l < 9; ++l) acc = fmaf(s[k][l], t[l], acc);
            o = fmaf(acc, dot[k], o);
        }
        out[cb] = o;   // NCHW, coalesced across lanes
    }
}

// ===========================================================================
extern "C" void kernel_launch(void* const* d_in, const int* in_sizes, int n_in,
                              void* d_out, int out_size, void* d_ws, size_t ws_size,
                              hipStream_t stream) {
    const float* x  = (const float*)d_in[0];
    const float* w1 = (const float*)d_in[1];
    const float* w2 = (const float*)d_in[2];
    const float* w3 = (const float*)d_in[3];

    float* y1 = (float*)d_ws;                 // 3 x N*C*H*W fp32 = 56.6 MB scratch
    float* y2 = y1 + (size_t)8 * IMG;
    float* y3 = y2 + (size_t)8 * IMG;

    // 73728 pixels / (8 waves * 16 pixels) = 576 blocks
    conv1x1x3_wmma_relu<<<576, 256, 0, stream>>>(x, w1, w2, w3, y1, y2, y3);
    // 73728 pixels / 256 threads = 288 blocks
    attention3x3<<<288, 256, 0, stream>>>(y1, y2, y3, (float*)d_out);
}